// RadiusAttentionWeights_44813688767184
// MI455X (gfx1250) — compile-verified
//
#include <hip/hip_runtime.h>

typedef float v2f __attribute__((ext_vector_type(2)));
typedef float v8f __attribute__((ext_vector_type(8)));

constexpr int   NPTS = 100000;
constexpr int   DIM  = 256;
constexpr int   NCEN = 2048;
constexpr int   KNN  = 32;
constexpr int   HID  = 1024;
constexpr float R2   = 23.0f * 23.0f;

// ds_swizzle_b32 group-of-32 XOR lane shuffle: offset = {0, xor_mask[4:0], or=0, and=0x1f}
template <int MASK>
__device__ __forceinline__ float swzf(float v) {
    return __int_as_float(__builtin_amdgcn_ds_swizzle(__float_as_int(v), (MASK << 10) | 0x1f));
}
template <int MASK>
__device__ __forceinline__ unsigned swzu(unsigned v) {
    return (unsigned)__builtin_amdgcn_ds_swizzle((int)v, (MASK << 10) | 0x1f);
}

// ---------------- init s / counts ----------------
__global__ void init_kernel(float* __restrict__ s, int* __restrict__ counts) {
    int i = blockIdx.x * blockDim.x + threadIdx.x;
    if (i < NCEN) { s[i] = 0.0f; counts[i] = 0; }
}

// ---------------- row norms nf[j] = |features[j]|^2 ----------------
__global__ void __launch_bounds__(256)
norm_kernel(const float* __restrict__ feat, float* __restrict__ nf) {
    int row  = blockIdx.x * 8 + (threadIdx.x >> 5);   // one wave32 per row
    int lane = threadIdx.x & 31;
    const float* p = feat + (size_t)row * DIM;
    float acc = 0.f;
    #pragma unroll
    for (int i = 0; i < DIM / 32; ++i) {
        float v = p[lane + 32 * i];                    // coalesced
        acc += v * v;
    }
    acc += swzf<16>(acc);
    acc += swzf<8>(acc);
    acc += swzf<4>(acc);
    acc += swzf<2>(acc);
    acc += swzf<1>(acc);
    if (lane == 0) nf[row] = acc;
}

// ---------------- MLP: s[c] += sum_h relu(x_c . W1[:,h] + b1[h]) * W2[h] ----------------
// one wave per 16(center) x 16(hidden) tile; 64x V_WMMA_F32_16X16X4_F32 over D=256
__global__ void __launch_bounds__(32)
mlp_kernel(const float* __restrict__ feat, const int* __restrict__ cidx,
           const float* __restrict__ W1,   const float* __restrict__ b1,
           const float* __restrict__ W2,   float* __restrict__ s) {
    const int cbase = blockIdx.x * 16;
    const int hbase = blockIdx.y * 16;
    const int lane  = threadIdx.x;           // 0..31
    const int n     = lane & 15;
    const int koff  = (lane < 16) ? 0 : 2;

    const float* xrow = feat + (size_t)cidx[cbase + n] * DIM + koff;     // A: M = lane%16
    const float* wcol = W1 + (size_t)koff * HID + hbase + n;             // B: N = lane%16

    v8f acc = {};
    #pragma unroll 8
    for (int d = 0; d < DIM; d += 4) {
        v2f a; a.x = xrow[d];                 a.y = xrow[d + 1];
        v2f b; b.x = wcol[(size_t)d * HID];   b.y = wcol[(size_t)d * HID + HID];
        acc = __builtin_amdgcn_wmma_f32_16x16x4_f32(false, a, false, b,
                                                    (short)0, acc, false, false);
    }

    const float bb = b1[hbase + n];
    const float w2 = W2[hbase + n];
    const int   mh = (lane >> 4) * 8;         // C/D: vgpr r -> M=r (lo half) / r+8 (hi half)
    #pragma unroll
    for (int r = 0; r < 8; ++r) {
        float z = acc[r] + bb;
        z = z > 0.f ? z : 0.f;
        z *= w2;
        z += swzf<1>(z);                      // reduce 16 N-lanes within each half
        z += swzf<2>(z);
        z += swzf<4>(z);
        z += swzf<8>(z);
        if (n == 0) atomicAdd(&s[cbase + mh + r], z);
    }
}

// ---------------- radius counts: #{ j : |x_c - x_j|^2 <= R^2 } per center ----------------
// block = 8 waves; 32 centers (LDS tile) x 512 features per block.
// Each wave register-blocks 2 center-tiles x 4 feature-tiles (8 accumulators):
// per k-step, 2 LDS dwordx2 + 4 global dwordx2 feed 8 v_wmma_f32_16x16x4_f32.
__global__ void __launch_bounds__(256)
dist_kernel(const float* __restrict__ feat, const int* __restrict__ cidx,
            const float* __restrict__ nf,   int* __restrict__ counts) {
    __shared__ float ldsC[32 * 260];          // 260-float row pitch -> banks spread
    __shared__ float ncS[32];
    __shared__ int   cntS[32];

    const int cbase = blockIdx.y * 32;
    const int tid   = threadIdx.x;

    { // stage 32x256 center tile (gather via center_idx)
        int row = tid >> 3;                   // 0..31
        int cg  = (tid & 7) * 32;             // 0..224
        const float* src = feat + (size_t)cidx[cbase + row] * DIM + cg;
        float* dst = ldsC + row * 260 + cg;
        #pragma unroll
        for (int i = 0; i < 32; i += 4)
            *(float4*)(dst + i) = *(const float4*)(src + i);
        if (tid < 32) { ncS[tid] = nf[cidx[cbase + tid]]; cntS[tid] = 0; }
    }
    __syncthreads();

    const int wave = tid >> 5;
    const int lane = tid & 31;
    const int n    = lane & 15;
    const int koff = (lane < 16) ? 0 : 2;
    const int j0   = blockIdx.x * 512 + wave * 64;

    int jcol[4], jj[4];
    const float* fp[4];
    float nfj[4];
    #pragma unroll
    for (int t = 0; t < 4; ++t) {
        jcol[t] = j0 + t * 16 + n;
        jj[t]   = (jcol[t] < NPTS) ? jcol[t] : (NPTS - 1);  // branchless clamp
        fp[t]   = feat + (size_t)jj[t] * DIM + koff;
        nfj[t]  = nf[jj[t]];
        __builtin_prefetch(fp[t], 0, 1);      // global_prefetch_b8 on the B rows
    }
    const float* ap0 = ldsC + n * 260 + koff;         // center tile 0 (M=0..15)
    const float* ap1 = ldsC + (n + 16) * 260 + koff;  // center tile 1 (M=16..31)

    v8f acc[8] = {};   // [cTile*4 + fTile]
    #pragma unroll 4
    for (int d = 0; d < DIM; d += 4) {
        v2f a0, a1, b[4];
        a0.x = ap0[d]; a0.y = ap0[d + 1];
        a1.x = ap1[d]; a1.y = ap1[d + 1];
        #pragma unroll
        for (int t = 0; t < 4; ++t) { b[t].x = fp[t][d]; b[t].y = fp[t][d + 1]; }
        #pragma unroll
        for (int t = 0; t < 4; ++t) {
            acc[t]     = __builtin_amdgcn_wmma_f32_16x16x4_f32(false, a0, false, b[t],
                                                               (short)0, acc[t],     false, false);
            acc[4 + t] = __builtin_amdgcn_wmma_f32_16x16x4_f32(false, a1, false, b[t],
                                                               (short)0, acc[4 + t], false, false);
        }
    }

    // Epilogue: byte-packed per-lane hit counts, xor-swizzle reduce across 16 N-lanes,
    // then 2 divergent lanes (0 and 16) commit 8 LDS atomics each per center-tile.
    const int mh = (lane >> 4) * 8;
    #pragma unroll
    for (int ct = 0; ct < 2; ++ct) {
        unsigned p0 = 0, p1 = 0;   // bytes: r=0..3 in p0, r=4..7 in p1 (each count <= 4)
        #pragma unroll
        for (int t = 0; t < 4; ++t) {
            bool jv = (jcol[t] < NPTS);
            #pragma unroll
            for (int r = 0; r < 8; ++r) {
                // d2 <= R^2  <=>  dot >= (nc + nf - R^2)/2
                float thr = 0.5f * (ncS[ct * 16 + mh + r] + nfj[t] - R2);
                unsigned hit = (jv && (acc[ct * 4 + t][r] >= thr)) ? 1u : 0u;
                if (r < 4) p0 += hit << (8 * r);
                else       p1 += hit << (8 * (r - 4));
            }
        }
        // byte-wise sums across 16 lanes: max 16*4 = 64 < 256, no carry between bytes
        p0 += swzu<1>(p0);  p1 += swzu<1>(p1);
        p0 += swzu<2>(p0);  p1 += swzu<2>(p1);
        p0 += swzu<4>(p0);  p1 += swzu<4>(p1);
        p0 += swzu<8>(p0);  p1 += swzu<8>(p1);
        if (n == 0) {   // lanes 0 (mh=0) and 16 (mh=8): disjoint cntS rows
            #pragma unroll
            for (int r = 0; r < 4; ++r) {
                int c0 = (int)((p0 >> (8 * r)) & 0xFFu);
                int c1 = (int)((p1 >> (8 * r)) & 0xFFu);
                if (c0) atomicAdd(&cntS[ct * 16 + mh + r], c0);
                if (c1) atomicAdd(&cntS[ct * 16 + mh + 4 + r], c1);
            }
        }
    }
    __syncthreads();
    if (tid < 32) {
        int v = cntS[tid];
        if (v) atomicAdd(&counts[cbase + tid], v);
    }
}

// ---------------- final: out[c*K+k] = (k < min(count,K)) ? s_c + b2 : 0 ----------------
__global__ void out_kernel(const float* __restrict__ s, const int* __restrict__ counts,
                           const float* __restrict__ b2, float* __restrict__ out) {
    int t = blockIdx.x * blockDim.x + threadIdx.x;
    if (t >= NCEN * KNN) return;
    int c = t >> 5;                 // KNN == 32
    int k = t & 31;
    int m = counts[c]; if (m > KNN) m = KNN;
    out[t] = (k < m) ? (s[c] + b2[0]) : 0.0f;
}

extern "C" void kernel_launch(void* const* d_in, const int* in_sizes, int n_in,
                              void* d_out, int out_size, void* d_ws, size_t ws_size,
                              hipStream_t stream) {
    const int*   cidx = (const int*)  d_in[0];
    const float* feat = (const float*)d_in[1];
    // d_in[2] = edge_index: unused by the reference computation
    const float* W1   = (const float*)d_in[3];
    const float* b1   = (const float*)d_in[4];
    const float* W2   = (const float*)d_in[5];
    const float* b2   = (const float*)d_in[6];
    float* out = (float*)d_out;

    // workspace layout
    float* nf     = (float*)d_ws;          // NPTS floats (row norms)
    float* s      = nf + 100352;           // NCEN floats (per-center MLP scalar)
    int*   counts = (int*)(s + NCEN);      // NCEN ints   (in-radius counts)

    init_kernel<<<(NCEN + 255) / 256, 256, 0, stream>>>(s, counts);
    norm_kernel<<<NPTS / 8, 256, 0, stream>>>(feat, nf);
    mlp_kernel <<<dim3(NCEN / 16, HID / 16), 32, 0, stream>>>(feat, cidx, W1, b1, W2, s);
    dist_kernel<<<dim3((NPTS + 511) / 512, NCEN / 32), 256, 0, stream>>>(feat, cidx, nf, counts);
    out_kernel <<<(NCEN * KNN + 255) / 256, 256, 0, stream>>>(s, counts, b2, out);
}